// UnPooling3D_74474732913174
// MI455X (gfx1250) — compile-verified
//
#include <hip/hip_runtime.h>
#include <hip/hip_bf16.h>
#include <stdint.h>

// UnPooling3D: in (4,32,32,32,64) f32 -> out (4,64,64,64,64) f32.
// out[b, 2d, 2h, 2w, :] = in[b, d, h, w, :]; everything else zero.
// Pure data movement: 32 MiB read + 256 MiB written -> HBM-store bound
// (~12.4 us at 23.3 TB/s). Strategy: b128 accesses, non-temporal hints on
// every store (output 256 MiB > 192 MiB L2, written once, never re-read),
// and the gfx1250 async LDS->global store path (ASYNCcnt) for the 7/8 of
// output bytes that are zeros.

typedef float __attribute__((ext_vector_type(4))) f4;
typedef int   __attribute__((ext_vector_type(4))) v4i;

#if defined(__gfx1250__) &&                                              \
    __has_builtin(__builtin_amdgcn_global_store_async_from_lds_b128) &&  \
    __has_builtin(__builtin_amdgcn_s_wait_asynccnt)
#define UNPOOL_ASYNC 1
#else
#define UNPOOL_ASYNC 0
#endif

#if UNPOOL_ASYNC
// Builtin signature (confirmed by clang diagnostics + successful codegen):
//   void __builtin_amdgcn_global_store_async_from_lds_b128(
//       int4 addrspace(1)* dst, int4 addrspace(3)* src, imm int off, imm int cpol)
typedef __attribute__((address_space(1))) v4i gv4i;  // global int4
typedef __attribute__((address_space(3))) v4i lv4i;  // LDS int4
// CPol: TH in bits [2:0]; TH_STORE_NT == 1 (stream past caches).
#define UNPOOL_CPOL_NT 1
#endif

__global__ __launch_bounds__(256) void unpool3d_kernel(
    const f4* __restrict__ in, f4* __restrict__ out)
{
    const int tid = threadIdx.x;
    const int blk = blockIdx.x;          // blk = ((b*64)+od)*64 + oh
    const int oh  = blk & 63;
    const int od  = (blk >> 6) & 63;
    const int b   = blk >> 12;

    // Output row (w=64, c=64) = 4096 floats = 1024 float4 = 16 KiB.
    f4* __restrict__ rowOut = out + ((long)blk << 10);

#if UNPOOL_ASYNC
    // 512 B of zeros in LDS, 16 B per lane; streamed out via the async
    // LDS->global store data mover (CDNA5-specific, tracked by ASYNCcnt).
    __shared__ __align__(16) float zbuf[128];
    if (tid < 128) zbuf[tid] = 0.0f;
    __syncthreads();
    const int lane = tid & 31;
    // Low 32 bits of a generic LDS pointer are the LDS byte offset.
    lv4i* zsrc = (lv4i*)(uint32_t)(uintptr_t)(&zbuf[lane * 4]);
#define ZSTORE(p)                                                         \
    __builtin_amdgcn_global_store_async_from_lds_b128(                    \
        (gv4i*)(uintptr_t)(p), zsrc, 0, UNPOOL_CPOL_NT)
#else
    const f4 zero = {0.f, 0.f, 0.f, 0.f};
#define ZSTORE(p) __builtin_nontemporal_store(zero, (f4*)(p))
#endif

    if ((od | oh) & 1) {
        // 75% of rows: pure zero fill, 4 b128 stores per thread.
        #pragma unroll
        for (int j = 0; j < 4; ++j) {
            ZSTORE(rowOut + tid + (j << 8));
        }
    } else {
        // Mixed row: even w -> copy input voxel float4, odd w -> zero.
        const int id = od >> 1, ih = oh >> 1;
        // Input row: 32 voxels * 16 float4 = 512 float4.
        const f4* __restrict__ inRow =
            in + ((long)((b * 32 + id) * 32 + ih) << 9);
        #pragma unroll
        for (int j = 0; j < 4; ++j) {
            const int f  = tid + (j << 8);   // 0..1023
            const int w  = f >> 4;
            const int c4 = f & 15;
            if (w & 1) {
                ZSTORE(rowOut + f);
            } else {
                // Input is read exactly once -> NT load; output streamed NT.
                f4 v = __builtin_nontemporal_load(inRow + ((w >> 1) << 4) + c4);
                __builtin_nontemporal_store(v, rowOut + f);
            }
        }
    }

#if UNPOOL_ASYNC
    // All async LDS->global stores must retire before wave exit.
    __builtin_amdgcn_s_wait_asynccnt(0);
#endif
#undef ZSTORE
}

extern "C" void kernel_launch(void* const* d_in, const int* in_sizes, int n_in,
                              void* d_out, int out_size, void* d_ws, size_t ws_size,
                              hipStream_t stream) {
    (void)in_sizes; (void)n_in; (void)out_size; (void)d_ws; (void)ws_size;
    const f4* in = (const f4*)d_in[0];
    f4* out = (f4*)d_out;
    // One block per output (b, od, oh) row: 4*64*64 = 16384 blocks.
    unpool3d_kernel<<<dim3(16384), dim3(256), 0, stream>>>(in, out);
}